// PGCN_h_64845416235269
// MI455X (gfx1250) — compile-verified
//
#include <hip/hip_runtime.h>
#include <math.h>

// ---------------- problem constants (match reference) ----------------
#define Bg   32
#define Nn   1024
#define Ff   128
#define DEG  16
#define Ecnt (Bg * Nn * DEG)   // 524288

typedef float v8f __attribute__((ext_vector_type(8)));
typedef float v2f __attribute__((ext_vector_type(2)));

// ---------------- generic fills ----------------
__global__ void fill_f32(float* p, float v, int n) {
  int i = blockIdx.x * blockDim.x + threadIdx.x;
  if (i < n) p[i] = v;
}

// ---------------- degree accumulation ----------------
// level 1: 4 edge-weight channels, edge_attr is [E,4] row-major
__global__ void deg_scatter4(const int* dst, const float* ea, float* deg, int E, int n0) {
  int idx = blockIdx.x * blockDim.x + threadIdx.x;
  if (idx >= 4 * E) return;
  int e = idx >> 2, c = idx & 3;
  atomicAdd(&deg[c * n0 + dst[e]], ea[e * 4 + c]);
}
// levels 2/3: weight = emask (0/1)
__global__ void deg_scatter_mask(const int* dst, const int* em, float* deg, int E) {
  int e = blockIdx.x * blockDim.x + threadIdx.x;
  if (e >= E) return;
  if (em[e]) atomicAdd(&deg[dst[e]], 1.0f);
}
__global__ void rsqrt_inplace(float* p, int n) {
  int i = blockIdx.x * blockDim.x + threadIdx.x;
  if (i < n) p[i] = rsqrtf(p[i]);   // deg >= 1 always (self loop)
}

// ---------------- pack 4 GCN weights into one [128,512] matrix ----------------
__global__ void pack_w4(const float* a, const float* b, const float* c, const float* d,
                        float* W4) {
  int idx = blockIdx.x * blockDim.x + threadIdx.x;   // 128*512
  if (idx >= 128 * 512) return;
  int row = idx >> 9, col = idx & 511;
  int ch = col >> 7, j = col & 127;
  const float* src = (ch == 0) ? a : (ch == 1) ? b : (ch == 2) ? c : d;
  W4[idx] = src[row * 128 + j];
}

// ---------------- fp32 WMMA GEMM: C[M,Nc] = A[M,K] @ B[K,Nc] (+bias)(+relu) ------
// One wave computes one 16x16 tile via V_WMMA_F32_16X16X4_F32.
// A lane layout (16x4 f32):  lane = 16*h + m15  holds A[m15][2h], A[m15][2h+1]
// B lane layout (4x16 f32):  lane = 16*h + n15  holds B[2h][n15], B[2h+1][n15]
// C/D: vgpr v, lane l -> C[v + 8*(l>>4)][l&15]
__global__ void gemm_wmma_f32(const float* __restrict__ A, const float* __restrict__ Bm,
                              float* __restrict__ C, int M, int K, int Nc,
                              const float* __restrict__ bias, int do_relu) {
  int gw   = (blockIdx.x * blockDim.x + threadIdx.x) >> 5;
  int lane = threadIdx.x & 31;
  int tn_cnt = Nc >> 4;
  int total  = (M >> 4) * tn_cnt;
  if (gw >= total) return;
  int tm = gw / tn_cnt, tn = gw - tm * tn_cnt;
  int half = lane >> 4, l15 = lane & 15;

  const float* arow = A + (long)(tm * 16 + l15) * K + 2 * half;
  const float* bptr = Bm + (long)(2 * half) * Nc + tn * 16 + l15;

  v8f acc = {};
  for (int k = 0; k < K; k += 4) {
    v2f av, bv;
    av.x = arow[k];
    av.y = arow[k + 1];
    bv.x = bptr[(long)k * Nc];
    bv.y = bptr[(long)(k + 1) * Nc];
    acc = __builtin_amdgcn_wmma_f32_16x16x4_f32(false, av, false, bv,
                                                (short)0, acc, false, false);
  }
  int col = tn * 16 + l15;
  float bb = bias ? bias[col] : 0.0f;
#pragma unroll
  for (int v = 0; v < 8; ++v) {
    float val = acc[v] + bb;
    if (do_relu) val = fmaxf(val, 0.0f);
    C[(long)(tm * 16 + v + 8 * half) * Nc + col] = val;
  }
}

// ---------------- GCN: self-loop term + bias, then edge scatter ----------------
__global__ void gcn_self_bias(const float* xw, const float* dis, const float* bias,
                              float* out, int n, int Fd, int ld) {
  int idx = blockIdx.x * blockDim.x + threadIdx.x;
  if (idx >= n * Fd) return;
  int i = idx / Fd, j = idx - i * Fd;
  float d = dis[i];
  out[(long)i * ld + j] = d * d * xw[(long)i * ld + j] + bias[j];
}

// one block (Fd threads) per edge; ew==nullptr => weight = 1 (mask in em)
__global__ void gcn_scatter(const float* __restrict__ xw, const float* __restrict__ dis,
                            const int* __restrict__ src, const int* __restrict__ dst,
                            const float* __restrict__ ew, int ewstride,
                            const int* __restrict__ em,
                            float* __restrict__ out, int E, int ld) {
  int e = blockIdx.x;
  if (e >= E) return;
  if (em && !em[e]) return;
  int s = src[e], d = dst[e];
  float w = ew ? ew[(long)e * ewstride] : 1.0f;
  float coef = dis[s] * w * dis[d];
  int j = threadIdx.x;
  atomicAdd(&out[(long)d * ld + j], coef * xw[(long)s * ld + j]);
}

__global__ void relu_inplace(float* p, int n) {
  int i = blockIdx.x * blockDim.x + threadIdx.x;
  if (i < n) p[i] = fmaxf(p[i], 0.0f);
}

// ---------------- GraphConv score: t = X.wrel, r = X.wroot (one wave/row) -------
__global__ void dual_dot(const float* __restrict__ X, int ld, int Kd,
                         const float* __restrict__ w1, const float* __restrict__ w2,
                         float* t, float* r, int n) {
  int wid  = (blockIdx.x * blockDim.x + threadIdx.x) >> 5;
  int lane = threadIdx.x & 31;
  if (wid >= n) return;
  const float* row = X + (long)wid * ld;
  float a = 0.0f, b = 0.0f;
  for (int j = lane; j < Kd; j += 32) {
    float xv = row[j];
    a += xv * w1[j];
    b += xv * w2[j];
  }
  for (int off = 16; off; off >>= 1) {
    a += __shfl_xor(a, off);
    b += __shfl_xor(b, off);
  }
  if (lane == 0) { t[wid] = a; r[wid] = b; }
}

__global__ void score_scatter(const int* src, const int* dst, const int* em,
                              const float* t, float* srel, int E) {
  int e = blockIdx.x * blockDim.x + threadIdx.x;
  if (e >= E) return;
  if (em && !em[e]) return;
  atomicAdd(&srel[dst[e]], t[src[e]]);
}

// score = tanh(srel + brel + r), in place in srel
__global__ void finish_score(float* srel, const float* r, const float* brel, int n) {
  int i = blockIdx.x * blockDim.x + threadIdx.x;
  if (i < n) srel[i] = tanhf(srel[i] + brel[0] + r[i]);
}

// ---------------- SAGPool top-k: per-graph LDS bitonic sort (descending) --------
// block = n_cur threads, grid = Bg graphs, shared = n_cur*8 bytes
__global__ void topk_pool(const float* score, int n_cur, int kk,
                          int* oldidx, float* tops, int* newid) {
  extern __shared__ char smraw[];
  float* sval = (float*)smraw;
  int*   sidx = (int*)(smraw + (size_t)n_cur * sizeof(float));
  int g = blockIdx.x, tid = threadIdx.x;
  sval[tid] = score[g * n_cur + tid];
  sidx[tid] = tid;
  newid[g * n_cur + tid] = -1;
  __syncthreads();
  for (int ksz = 2; ksz <= n_cur; ksz <<= 1) {
    for (int j = ksz >> 1; j > 0; j >>= 1) {
      int ixj = tid ^ j;
      if (ixj > tid) {
        bool desc = ((tid & ksz) == 0);
        float a = sval[tid], b = sval[ixj];
        bool sw = desc ? (a < b) : (a > b);
        if (sw) {
          sval[tid] = b; sval[ixj] = a;
          int ti = sidx[tid]; sidx[tid] = sidx[ixj]; sidx[ixj] = ti;
        }
      }
      __syncthreads();
    }
  }
  if (tid < kk) {
    int node = sidx[tid];
    oldidx[g * kk + tid] = g * n_cur + node;
    tops[g * kk + tid]   = sval[tid];
    newid[g * n_cur + node] = g * kk + tid;
  }
}

// x_new[row] = x[old[row]] * tops[row]; block = Fd threads, grid = B*kk
__global__ void gather_scale(const float* __restrict__ xin, int Fd,
                             const int* __restrict__ oldidx, const float* __restrict__ tops,
                             float* __restrict__ xout) {
  int row = blockIdx.x, j = threadIdx.x;
  float s = tops[row];
  xout[(long)row * Fd + j] = xin[(long)oldidx[row] * Fd + j] * s;
}

__global__ void remap_edges(const int* src, const int* dst, const int* em_in,
                            const int* newid, int* s_out, int* d_out, int* em_out, int E) {
  int e = blockIdx.x * blockDim.x + threadIdx.x;
  if (e >= E) return;
  int s2 = newid[src[e]], d2 = newid[dst[e]];
  int ok = (em_in ? em_in[e] : 1) && (s2 >= 0) && (d2 >= 0);
  em_out[e] = ok;
  s_out[e]  = ok ? s2 : 0;
  d_out[e]  = ok ? d2 : 0;
}

// ---------------- readout: mean & max over kk nodes per graph ----------------
__global__ void readout(const float* __restrict__ x, int kk, int Fd,
                        float* __restrict__ r, int ldr, int coloff, int total) {
  int idx = blockIdx.x * blockDim.x + threadIdx.x;
  if (idx >= total) return;
  int g = idx / Fd, f = idx - g * Fd;
  const float* p = x + ((long)g * kk) * Fd + f;
  float s = 0.0f, m = -INFINITY;
  for (int t = 0; t < kk; ++t) {
    float v = p[(long)t * Fd];
    s += v;
    m = fmaxf(m, v);
  }
  r[(long)g * ldr + coloff + f]      = s / (float)kk;
  r[(long)g * ldr + coloff + Fd + f] = m;
}

// ---------------- tiny dense layers (MLP tail) ----------------
__global__ void linear_small(const float* X, const float* W, const float* bias,
                             float* out, int M, int Kd, int Nc, int do_relu) {
  int idx = blockIdx.x * blockDim.x + threadIdx.x;
  if (idx >= M * Nc) return;
  int m = idx / Nc, nn = idx - m * Nc;
  float s = bias[nn];
  for (int k = 0; k < Kd; ++k) s += X[m * Kd + k] * W[k * Nc + nn];
  if (do_relu) s = fmaxf(s, 0.0f);
  out[idx] = s;
}

// =====================================================================
extern "C" void kernel_launch(void* const* d_in, const int* in_sizes, int n_in,
                              void* d_out, int out_size, void* d_ws, size_t ws_size,
                              hipStream_t stream) {
  (void)in_sizes; (void)n_in; (void)out_size; (void)ws_size;
  const int n0 = Bg * Nn;          // 32768
  const int k1 = Nn / 2,  n1 = Bg * k1;   // 512, 16384
  const int k2 = k1 / 2,  n2 = Bg * k2;   // 256, 8192
  const int k3 = k2 / 2;                  // 128
  const int E  = Ecnt;

  // ---- inputs (setup_inputs order) ----
  const float* x    = (const float*)d_in[0];
  const int*   src0 = (const int*)d_in[1];
  const int*   dst0 = src0 + E;
  const float* ea   = (const float*)d_in[2];
  // d_in[3] = batch (unused)
  const float* W1[4] = {(const float*)d_in[4], (const float*)d_in[6],
                        (const float*)d_in[8], (const float*)d_in[10]};
  const float* b1[4] = {(const float*)d_in[5], (const float*)d_in[7],
                        (const float*)d_in[9], (const float*)d_in[11]};
  const float* p1_wrel  = (const float*)d_in[12];
  const float* p1_brel  = (const float*)d_in[13];
  const float* p1_wroot = (const float*)d_in[14];
  const float* W2 = (const float*)d_in[15];
  const float* b2 = (const float*)d_in[16];
  const float* p2_wrel  = (const float*)d_in[17];
  const float* p2_brel  = (const float*)d_in[18];
  const float* p2_wroot = (const float*)d_in[19];
  const float* W3 = (const float*)d_in[20];
  const float* b3 = (const float*)d_in[21];
  const float* p3_wrel  = (const float*)d_in[22];
  const float* p3_brel  = (const float*)d_in[23];
  const float* p3_wroot = (const float*)d_in[24];
  const float* mw1 = (const float*)d_in[25];
  const float* mb1 = (const float*)d_in[26];
  const float* mw2 = (const float*)d_in[27];
  const float* mb2 = (const float*)d_in[28];
  const float* mw3 = (const float*)d_in[29];
  const float* mb3 = (const float*)d_in[30];

  // ---- workspace carve (256B aligned bump) ----
  char* wsb = (char*)d_ws;
  auto alloc = [&](size_t bytes) -> void* {
    void* p = (void*)wsb;
    wsb += (bytes + 255) & ~(size_t)255;
    return p;
  };
  float* W4    = (float*)alloc(sizeof(float) * 128 * 512);
  float* dis4  = (float*)alloc(sizeof(float) * 4 * n0);      // deg -> rsqrt in place
  float* xw4   = (float*)alloc(sizeof(float) * (size_t)n0 * 512);
  float* gcn1  = (float*)alloc(sizeof(float) * (size_t)n0 * 512);
  float* t1    = (float*)alloc(sizeof(float) * n0);
  float* r1    = (float*)alloc(sizeof(float) * n0);
  float* srel1 = (float*)alloc(sizeof(float) * n0);
  int*   old1  = (int*)alloc(sizeof(int) * Bg * k1);
  float* tops1 = (float*)alloc(sizeof(float) * Bg * k1);
  int*   newid = (int*)alloc(sizeof(int) * n0);
  float* p1x   = (float*)alloc(sizeof(float) * (size_t)n1 * 512);
  int*   s1    = (int*)alloc(sizeof(int) * E);
  int*   d1    = (int*)alloc(sizeof(int) * E);
  int*   em1   = (int*)alloc(sizeof(int) * E);
  float* dis2  = (float*)alloc(sizeof(float) * n1);
  float* xw2   = (float*)alloc(sizeof(float) * (size_t)n1 * 256);
  float* gcn2  = (float*)alloc(sizeof(float) * (size_t)n1 * 256);
  float* t2    = (float*)alloc(sizeof(float) * n1);
  float* r2    = (float*)alloc(sizeof(float) * n1);
  float* srel2 = (float*)alloc(sizeof(float) * n1);
  int*   old2  = (int*)alloc(sizeof(int) * Bg * k2);
  float* tops2 = (float*)alloc(sizeof(float) * Bg * k2);
  float* p2x   = (float*)alloc(sizeof(float) * (size_t)n2 * 256);
  int*   s2    = (int*)alloc(sizeof(int) * E);
  int*   d2    = (int*)alloc(sizeof(int) * E);
  int*   em2   = (int*)alloc(sizeof(int) * E);
  float* dis3  = (float*)alloc(sizeof(float) * n2);
  float* xw3   = (float*)alloc(sizeof(float) * (size_t)n2 * 128);
  float* gcn3  = (float*)alloc(sizeof(float) * (size_t)n2 * 128);
  float* t3    = (float*)alloc(sizeof(float) * n2);
  float* r3    = (float*)alloc(sizeof(float) * n2);
  float* srel3 = (float*)alloc(sizeof(float) * n2);
  int*   old3  = (int*)alloc(sizeof(int) * Bg * k3);
  float* tops3 = (float*)alloc(sizeof(float) * Bg * k3);
  float* p3x   = (float*)alloc(sizeof(float) * (size_t)Bg * k3 * 128);
  float* rbuf  = (float*)alloc(sizeof(float) * Bg * 1792);
  float* h1    = (float*)alloc(sizeof(float) * Bg * 32);
  float* h2    = (float*)alloc(sizeof(float) * Bg * 8);

  auto cdiv = [](int a, int b) { return (a + b - 1) / b; };
  auto gemm = [&](const float* A, const float* Bm, float* C, int M, int K, int Nc,
                  const float* bias, int relu) {
    int tiles = (M / 16) * (Nc / 16);
    gemm_wmma_f32<<<cdiv(tiles, 4), 128, 0, stream>>>(A, Bm, C, M, K, Nc, bias, relu);
  };

  // ================= level 1: 4x GCN on x =================
  fill_f32<<<cdiv(4 * n0, 256), 256, 0, stream>>>(dis4, 1.0f, 4 * n0);   // +1 self loop
  deg_scatter4<<<cdiv(4 * E, 256), 256, 0, stream>>>(dst0, ea, dis4, E, n0);
  rsqrt_inplace<<<cdiv(4 * n0, 256), 256, 0, stream>>>(dis4, 4 * n0);
  pack_w4<<<cdiv(128 * 512, 256), 256, 0, stream>>>(W1[0], W1[1], W1[2], W1[3], W4);
  gemm(x, W4, xw4, n0, 128, 512, nullptr, 0);                 // xw4 = x @ [W1A|W1B|W1C|W1D]
  for (int c = 0; c < 4; ++c) {
    gcn_self_bias<<<cdiv(n0 * 128, 256), 256, 0, stream>>>(
        xw4 + c * 128, dis4 + c * n0, b1[c], gcn1 + c * 128, n0, 128, 512);
    gcn_scatter<<<E, 128, 0, stream>>>(xw4 + c * 128, dis4 + c * n0, src0, dst0,
                                       ea + c, 4, nullptr, gcn1 + c * 128, E, 512);
  }
  relu_inplace<<<cdiv(n0 * 512, 256), 256, 0, stream>>>(gcn1, n0 * 512);

  // score 1 + pool 1
  dual_dot<<<cdiv(n0 * 32, 256), 256, 0, stream>>>(gcn1, 512, 512, p1_wrel, p1_wroot, t1, r1, n0);
  fill_f32<<<cdiv(n0, 256), 256, 0, stream>>>(srel1, 0.0f, n0);
  score_scatter<<<cdiv(E, 256), 256, 0, stream>>>(src0, dst0, nullptr, t1, srel1, E);
  finish_score<<<cdiv(n0, 256), 256, 0, stream>>>(srel1, r1, p1_brel, n0);
  topk_pool<<<Bg, Nn, Nn * 8, stream>>>(srel1, Nn, k1, old1, tops1, newid);
  gather_scale<<<n1, 512, 0, stream>>>(gcn1, 512, old1, tops1, p1x);
  remap_edges<<<cdiv(E, 256), 256, 0, stream>>>(src0, dst0, nullptr, newid, s1, d1, em1, E);
  readout<<<cdiv(Bg * 512, 256), 256, 0, stream>>>(p1x, k1, 512, rbuf, 1792, 0, Bg * 512);

  // ================= level 2 =================
  fill_f32<<<cdiv(n1, 256), 256, 0, stream>>>(dis2, 1.0f, n1);
  deg_scatter_mask<<<cdiv(E, 256), 256, 0, stream>>>(d1, em1, dis2, E);
  rsqrt_inplace<<<cdiv(n1, 256), 256, 0, stream>>>(dis2, n1);
  gemm(p1x, W2, xw2, n1, 512, 256, nullptr, 0);
  gcn_self_bias<<<cdiv(n1 * 256, 256), 256, 0, stream>>>(xw2, dis2, b2, gcn2, n1, 256, 256);
  gcn_scatter<<<E, 256, 0, stream>>>(xw2, dis2, s1, d1, nullptr, 0, em1, gcn2, E, 256);
  relu_inplace<<<cdiv(n1 * 256, 256), 256, 0, stream>>>(gcn2, n1 * 256);

  dual_dot<<<cdiv(n1 * 32, 256), 256, 0, stream>>>(gcn2, 256, 256, p2_wrel, p2_wroot, t2, r2, n1);
  fill_f32<<<cdiv(n1, 256), 256, 0, stream>>>(srel2, 0.0f, n1);
  score_scatter<<<cdiv(E, 256), 256, 0, stream>>>(s1, d1, em1, t2, srel2, E);
  finish_score<<<cdiv(n1, 256), 256, 0, stream>>>(srel2, r2, p2_brel, n1);
  topk_pool<<<Bg, k1, k1 * 8, stream>>>(srel2, k1, k2, old2, tops2, newid);
  gather_scale<<<n2, 256, 0, stream>>>(gcn2, 256, old2, tops2, p2x);
  remap_edges<<<cdiv(E, 256), 256, 0, stream>>>(s1, d1, em1, newid, s2, d2, em2, E);
  readout<<<cdiv(Bg * 256, 256), 256, 0, stream>>>(p2x, k2, 256, rbuf, 1792, 1024, Bg * 256);

  // ================= level 3 (no relu on gcn3) =================
  fill_f32<<<cdiv(n2, 256), 256, 0, stream>>>(dis3, 1.0f, n2);
  deg_scatter_mask<<<cdiv(E, 256), 256, 0, stream>>>(d2, em2, dis3, E);
  rsqrt_inplace<<<cdiv(n2, 256), 256, 0, stream>>>(dis3, n2);
  gemm(p2x, W3, xw3, n2, 256, 128, nullptr, 0);
  gcn_self_bias<<<cdiv(n2 * 128, 256), 256, 0, stream>>>(xw3, dis3, b3, gcn3, n2, 128, 128);
  gcn_scatter<<<E, 128, 0, stream>>>(xw3, dis3, s2, d2, nullptr, 0, em2, gcn3, E, 128);

  dual_dot<<<cdiv(n2 * 32, 256), 256, 0, stream>>>(gcn3, 128, 128, p3_wrel, p3_wroot, t3, r3, n2);
  fill_f32<<<cdiv(n2, 256), 256, 0, stream>>>(srel3, 0.0f, n2);
  score_scatter<<<cdiv(E, 256), 256, 0, stream>>>(s2, d2, em2, t3, srel3, E);
  finish_score<<<cdiv(n2, 256), 256, 0, stream>>>(srel3, r3, p3_brel, n2);
  topk_pool<<<Bg, k2, k2 * 8, stream>>>(srel3, k2, k3, old3, tops3, newid);
  gather_scale<<<Bg * k3, 128, 0, stream>>>(gcn3, 128, old3, tops3, p3x);
  readout<<<cdiv(Bg * 128, 256), 256, 0, stream>>>(p3x, k3, 128, rbuf, 1792, 1536, Bg * 128);

  // ================= MLP head =================
  gemm(rbuf, mw1, h1, Bg, 1792, 32, mb1, 1);   // WMMA, fused bias+relu
  linear_small<<<cdiv(Bg * 8, 64), 64, 0, stream>>>(h1, mw2, mb2, h2, Bg, 32, 8, 1);
  linear_small<<<cdiv(Bg * 2, 64), 64, 0, stream>>>(h2, mw3, mb3, (float*)d_out, Bg, 8, 2, 0);
}